// GATModel_78623671320995
// MI455X (gfx1250) — compile-verified
//
#include <hip/hip_runtime.h>
#include <hip/hip_bf16.h>
#include <cmath>

typedef float v2f __attribute__((ext_vector_type(2)));
typedef float v8f __attribute__((ext_vector_type(8)));

#define NEG_SLOPE 0.2f

// ---------------- utility fills ----------------
__global__ void fill_f32(float* __restrict__ p, float v, int n) {
  int i = blockIdx.x * blockDim.x + threadIdx.x;
  if (i < n) p[i] = v;
}

// out[n,128] = bias[c]  (fold +b0 into the layer-0 accumulator init)
__global__ void bias_init(float* __restrict__ xout, const float* __restrict__ bias, int N) {
  int i = blockIdx.x * blockDim.x + threadIdx.x;
  if (i < N * 128) xout[i] = bias[i & 127];
}

// ---------------- WMMA f32 GEMM ----------------
// Out[N,128] = A[N,Kdim] @ W[Kdim,128], A = rows < splitRow from A0 else A1.
// One 16x16 output tile per wave; 8 waves/block cover the full 128-col width
// of one 16-row stripe. K-loop in steps of 4 via V_WMMA_F32_16X16X4_F32.
__global__ __launch_bounds__(256)
void gemm_wmma(const float* __restrict__ A0, const float* __restrict__ A1,
               int splitRow, int N, int Kdim,
               const float* __restrict__ W, float* __restrict__ Out) {
  const int NC = 128;
  int wave = threadIdx.x >> 5;
  int lane = threadIdx.x & 31;
  int row0 = blockIdx.x << 4;
  int col0 = wave << 4;
  int m    = lane & 15;
  int half = lane >> 4;           // 0: K=k+0,k+1   1: K=k+2,k+3  (ISA 7.12.2)

  int arow = row0 + m;
  if (arow >= N) arow = N - 1;    // clamp (N is a multiple of 16 here anyway)
  const float* Arow = (arow < splitRow) ? (A0 + (size_t)arow * Kdim)
                                        : (A1 + (size_t)(arow - splitRow) * Kdim);
  int bcol = col0 + m;

  v8f acc = {};
  for (int k = 0; k < Kdim; k += 4) {
    int ka = k + 2 * half;
    v2f a, b;
    a.x = Arow[ka];
    a.y = Arow[ka + 1];
    b.x = W[(size_t)ka * NC + bcol];
    b.y = W[(size_t)(ka + 1) * NC + bcol];
    acc = __builtin_amdgcn_wmma_f32_16x16x4_f32(false, a, false, b,
                                                (short)0, acc, false, false);
  }
  // D layout: VGPR i -> row (half*8 + i), col (lane&15)
  int mb = half << 3;
#pragma unroll
  for (int i = 0; i < 8; ++i) {
    int r = row0 + mb + i;
    if (r < N) Out[(size_t)r * NC + bcol] = acc[i];
  }
}

// ---------------- attention scores ----------------
// a_src[n,h] = sum_c h[n,h,c]*att_src[h,c]; wave per node, shuffle reduce.
__global__ void att_scores(const float* __restrict__ Hm,
                           const float* __restrict__ asw, const float* __restrict__ adw,
                           float* __restrict__ a_src, float* __restrict__ a_dst, int N) {
  int wid  = (blockIdx.x * (blockDim.x >> 5)) + (threadIdx.x >> 5);
  int lane = threadIdx.x & 31;
  if (wid >= N) return;
  const float* hr = Hm + (size_t)wid * 128;
  float s0 = 0.f, s1 = 0.f, d0 = 0.f, d1 = 0.f;
#pragma unroll
  for (int i = 0; i < 2; ++i) {
    int c = lane + (i << 5);
    float h0 = hr[c], h1 = hr[64 + c];
    s0 += h0 * asw[c];      d0 += h0 * adw[c];
    s1 += h1 * asw[64 + c]; d1 += h1 * adw[64 + c];
  }
#pragma unroll
  for (int off = 16; off > 0; off >>= 1) {
    s0 += __shfl_down(s0, off, 32);
    s1 += __shfl_down(s1, off, 32);
    d0 += __shfl_down(d0, off, 32);
    d1 += __shfl_down(d1, off, 32);
  }
  if (lane == 0) {
    a_src[wid * 2]     = s0;  a_src[wid * 2 + 1] = s1;
    a_dst[wid * 2]     = d0;  a_dst[wid * 2 + 1] = d1;
  }
}

// ---------------- edge pass 1: logits + segment max ----------------
__global__ void edge_logits(const int* __restrict__ src, const int* __restrict__ dst,
                            const float* __restrict__ a_src, const float* __restrict__ a_dst,
                            float* __restrict__ ebuf, float* __restrict__ emax, int E) {
  int e = blockIdx.x * blockDim.x + threadIdx.x;
  if (e >= E) return;
  int s = src[e], d = dst[e];
#pragma unroll
  for (int h = 0; h < 2; ++h) {
    float v = a_src[s * 2 + h] + a_dst[d * 2 + h];
    v = (v > 0.f) ? v : NEG_SLOPE * v;
    ebuf[(size_t)e * 2 + h] = v;
    atomicMax(&emax[d * 2 + h], v);   // global_atomic_max_num_f32
  }
}

__global__ void emax_fix(float* __restrict__ emax, int n) {
  int i = blockIdx.x * blockDim.x + threadIdx.x;
  if (i < n) {
    float v = emax[i];
    if (!__builtin_isfinite(v)) emax[i] = 0.f;
  }
}

// ---------------- edge pass 2: exp + segment sum ----------------
__global__ void edge_exp(const int* __restrict__ dst, float* __restrict__ ebuf,
                         const float* __restrict__ emax, float* __restrict__ den, int E) {
  int e = blockIdx.x * blockDim.x + threadIdx.x;
  if (e >= E) return;
  int d = dst[e];
#pragma unroll
  for (int h = 0; h < 2; ++h) {
    float x = __expf(ebuf[(size_t)e * 2 + h] - emax[d * 2 + h]);
    ebuf[(size_t)e * 2 + h] = x;
    atomicAdd(&den[d * 2 + h], x);
  }
}

// ---------------- edge pass 3: weighted message scatter ----------------
// Wave per edge: 128 coalesced loads of h[src], 128 f32 atomic adds to out[dst].
__global__ void edge_scatter(const int* __restrict__ src, const int* __restrict__ dst,
                             const float* __restrict__ Hm, const float* __restrict__ ebuf,
                             const float* __restrict__ den, float* __restrict__ xout, int E) {
  int e    = (blockIdx.x * (blockDim.x >> 5)) + (threadIdx.x >> 5);
  int lane = threadIdx.x & 31;
  if (e >= E) return;
  int s = src[e], d = dst[e];
  float al0 = ebuf[(size_t)e * 2]     / (den[d * 2]     + 1e-16f);
  float al1 = ebuf[(size_t)e * 2 + 1] / (den[d * 2 + 1] + 1e-16f);
  const float* hr = Hm + (size_t)s * 128;
  float* orow = xout + (size_t)d * 128;
#pragma unroll
  for (int i = 0; i < 4; ++i) {
    int c = lane + (i << 5);
    float alpha = (c < 64) ? al0 : al1;
    atomicAdd(&orow[c], hr[c] * alpha);
  }
}

// ---------------- layer-1 epilogue: mean over heads + bias ----------------
__global__ void head_mean(const float* __restrict__ acc, const float* __restrict__ b1,
                          float* __restrict__ x2, int N) {
  int i = blockIdx.x * blockDim.x + threadIdx.x;
  if (i >= N * 64) return;
  int n = i >> 6, c = i & 63;
  x2[i] = 0.5f * (acc[(size_t)n * 128 + c] + acc[(size_t)n * 128 + 64 + c]) + b1[c];
}

// ---------------- final pair dot products ----------------
__global__ void pair_dot(const float* __restrict__ x2, const int* __restrict__ uidx,
                         const int* __restrict__ iidx, int NU,
                         float* __restrict__ out, int B) {
  int wid  = (blockIdx.x * (blockDim.x >> 5)) + (threadIdx.x >> 5);
  int lane = threadIdx.x & 31;
  if (wid >= B) return;
  const float* gu = x2 + (size_t)uidx[wid] * 64;
  const float* gi = x2 + (size_t)(NU + iidx[wid]) * 64;
  float s = gu[lane] * gi[lane] + gu[lane + 32] * gi[lane + 32];
#pragma unroll
  for (int off = 16; off > 0; off >>= 1) s += __shfl_down(s, off, 32);
  if (lane == 0) out[wid] = s;
}

extern "C" void kernel_launch(void* const* d_in, const int* in_sizes, int n_in,
                              void* d_out, int out_size, void* d_ws, size_t ws_size,
                              hipStream_t stream) {
  const float* Gu       = (const float*)d_in[0];
  const float* Gi       = (const float*)d_in[1];
  const float* W0       = (const float*)d_in[2];
  const float* att_src0 = (const float*)d_in[3];
  const float* att_dst0 = (const float*)d_in[4];
  const float* b0       = (const float*)d_in[5];
  const float* W1       = (const float*)d_in[6];
  const float* att_src1 = (const float*)d_in[7];
  const float* att_dst1 = (const float*)d_in[8];
  const float* b1       = (const float*)d_in[9];
  const int*   ei       = (const int*)d_in[10];
  const int*   uidx     = (const int*)d_in[11];
  const int*   iidx     = (const int*)d_in[12];
  float*       out      = (float*)d_out;

  const int D  = 64;
  int NU = in_sizes[0] / D;
  int NI = in_sizes[1] / D;
  int N  = NU + NI;
  int E  = in_sizes[10] / 2;
  int B  = in_sizes[11];
  const int* esrc = ei;       // edge_index[0]
  const int* edst = ei + E;   // edge_index[1]

  // workspace layout (all f32)
  float* w = (float*)d_ws;
  size_t off = 0;
  float* hbuf = w + off; off += (size_t)N * 128;  // h = x@W
  float* xacc = w + off; off += (size_t)N * 128;  // layer output accumulator
  float* x2   = w + off; off += (size_t)N * 64;   // layer-1 final features
  float* asrc = w + off; off += (size_t)N * 2;
  float* adst = w + off; off += (size_t)N * 2;
  float* emax = w + off; off += (size_t)N * 2;
  float* den  = w + off; off += (size_t)N * 2;
  float* ebuf = w + off; off += (size_t)E * 2;    // edge logits -> exp

  dim3 b256(256);
  auto nb = [](long n) { return dim3((unsigned)((n + 255) / 256)); };
  dim3 gemmGrid((N + 15) / 16);
  dim3 waveN((N + 7) / 8);   // 8 waves per 256-thread block
  dim3 waveE((E + 7) / 8);
  dim3 waveB((B + 7) / 8);

  // ---------------- layer 0 (concat=True, bias b0[128]) ----------------
  fill_f32 <<<nb(2L * N), b256, 0, stream>>>(emax, -INFINITY, 2 * N);
  fill_f32 <<<nb(2L * N), b256, 0, stream>>>(den, 0.f, 2 * N);
  bias_init<<<nb((long)N * 128), b256, 0, stream>>>(xacc, b0, N);
  gemm_wmma<<<gemmGrid, b256, 0, stream>>>(Gu, Gi, NU, N, 64, W0, hbuf);
  att_scores<<<waveN, b256, 0, stream>>>(hbuf, att_src0, att_dst0, asrc, adst, N);
  edge_logits<<<nb(E), b256, 0, stream>>>(esrc, edst, asrc, adst, ebuf, emax, E);
  emax_fix <<<nb(2L * N), b256, 0, stream>>>(emax, 2 * N);
  edge_exp <<<nb(E), b256, 0, stream>>>(edst, ebuf, emax, den, E);
  edge_scatter<<<waveE, b256, 0, stream>>>(esrc, edst, hbuf, ebuf, den, xacc, E);

  // ---------------- layer 1 (concat=False, bias b1[64]) ----------------
  gemm_wmma<<<gemmGrid, b256, 0, stream>>>(xacc, xacc, N, N, 128, W1, hbuf);
  fill_f32 <<<nb(2L * N), b256, 0, stream>>>(emax, -INFINITY, 2 * N);
  fill_f32 <<<nb(2L * N), b256, 0, stream>>>(den, 0.f, 2 * N);
  fill_f32 <<<nb((long)N * 128), b256, 0, stream>>>(xacc, 0.f, N * 128);
  att_scores<<<waveN, b256, 0, stream>>>(hbuf, att_src1, att_dst1, asrc, adst, N);
  edge_logits<<<nb(E), b256, 0, stream>>>(esrc, edst, asrc, adst, ebuf, emax, E);
  emax_fix <<<nb(2L * N), b256, 0, stream>>>(emax, 2 * N);
  edge_exp <<<nb(E), b256, 0, stream>>>(edst, ebuf, emax, den, E);
  edge_scatter<<<waveE, b256, 0, stream>>>(esrc, edst, hbuf, ebuf, den, xacc, E);
  head_mean<<<nb((long)N * 64), b256, 0, stream>>>(xacc, b1, x2, N);

  // ---------------- final scoring ----------------
  pair_dot<<<waveB, b256, 0, stream>>>(x2, uidx, iidx, NU, out, B);
}